// FP_Layer_11123965297224
// MI455X (gfx1250) — compile-verified
//
#include <hip/hip_runtime.h>

typedef float v2f __attribute__((ext_vector_type(2)));
typedef float v8f __attribute__((ext_vector_type(8)));

constexpr int kB    = 4;
constexpr int kNL   = 8192;
constexpr int kNH   = 2048;
constexpr int kCH   = 256;   // high-res feature channels (interpolated)
constexpr int kCL   = 128;   // low-res feature channels
constexpr int kCIN  = 384;   // kCH + kCL
constexpr int kCOUT = 256;
constexpr float kBnEps   = 1e-5f;
constexpr float kDistMin = 1e-8f;

// --------------------------------------------------------------------------
// Kernel 1: 3-NN + inverse-distance interpolation.
// Block = 256 threads (8 wave32), one wave per low point. xyz_high in LDS.
// Output finterp layout: (B, NL, CH)  -> coalesced lane-per-channel stores.
// --------------------------------------------------------------------------
__global__ __launch_bounds__(256) void knn_interp_kernel(
    const float* __restrict__ xyz_low,    // (B, NL, 3)
    const float* __restrict__ xyz_high,   // (B, NH, 3)
    const float* __restrict__ feat_high,  // (B, CH, NH)
    float* __restrict__ finterp)          // (B, NL, CH)
{
  __shared__ float sxyz[kNH * 3];
  const int b   = blockIdx.y;
  const int tid = threadIdx.x;

  for (int i = tid; i < kNH * 3; i += 256)
    sxyz[i] = xyz_high[(size_t)b * kNH * 3 + i];
  __syncthreads();

  const int wave = tid >> 5;
  const int lane = tid & 31;
  const int l    = blockIdx.x * 8 + wave;

  const float px = xyz_low[((size_t)b * kNL + l) * 3 + 0];
  const float py = xyz_low[((size_t)b * kNL + l) * 3 + 1];
  const float pz = xyz_low[((size_t)b * kNL + l) * 3 + 2];

  // per-lane sorted top-3 over its 64 candidates (squared distances)
  float bd0 = 1e30f, bd1 = 1e30f, bd2 = 1e30f;
  int   bi0 = -1,    bi1 = -1,    bi2 = -1;
  for (int j = lane; j < kNH; j += 32) {
    float dx = px - sxyz[3 * j + 0];
    float dy = py - sxyz[3 * j + 1];
    float dz = pz - sxyz[3 * j + 2];
    float d2 = fmaf(dx, dx, fmaf(dy, dy, dz * dz));
    if (d2 < bd2) {
      if (d2 < bd0)      { bd2=bd1; bi2=bi1; bd1=bd0; bi1=bi0; bd0=d2; bi0=j; }
      else if (d2 < bd1) { bd2=bd1; bi2=bi1; bd1=d2;  bi1=j; }
      else               { bd2=d2;  bi2=j; }
    }
  }

  // 3 rounds of wave-wide argmin (wave32 shuffles); every lane ends with
  // identical (idx, weight) triples.
  float wk[3]; int ik[3]; float wsum = 0.f;
  for (int k = 0; k < 3; ++k) {
    float v  = bd0;
    int  vid = bi0;
    for (int off = 16; off > 0; off >>= 1) {
      float ov  = __shfl_xor(v,  off, 32);
      int   oid = __shfl_xor(vid, off, 32);
      if (ov < v || (ov == v && oid < vid)) { v = ov; vid = oid; }
    }
    if (bi0 == vid) {   // global indices are unique -> exactly one lane pops
      bd0 = bd1; bi0 = bi1; bd1 = bd2; bi1 = bi2; bd2 = 1e30f; bi2 = -1;
    }
    float dist = fmaxf(sqrtf(fmaxf(v, 0.f)), kDistMin);
    wk[k] = 1.f / dist;
    ik[k] = vid;
    wsum += wk[k];
  }
  const float inv = 1.f / wsum;
  const float w0 = wk[0] * inv, w1 = wk[1] * inv, w2 = wk[2] * inv;
  const int   i0 = ik[0],       i1 = ik[1],       i2 = ik[2];

  const float* fh = feat_high + (size_t)b * kCH * kNH;
  float* dst = finterp + ((size_t)b * kNL + l) * kCH;
  for (int c = lane; c < kCH; c += 32) {
    float acc = w0 * fh[(size_t)c * kNH + i0];
    acc = fmaf(w1, fh[(size_t)c * kNH + i1], acc);
    acc = fmaf(w2, fh[(size_t)c * kNH + i2], acc);
    dst[c] = acc;
  }
}

// --------------------------------------------------------------------------
// Kernel 2: y = W @ f_cat via V_WMMA_F32_16X16X4_F32 (fp32 matrix pipe).
// Block = 8 waves; each wave owns one 16x16 output tile: rows
// mt*16..mt*16+15, cols n0..n0+15. f_cat(384 x 16) tile staged transposed in
// LDS with stride 388 (388 % 64 == 4 -> conflict-free float2 reads).
// Conv bias is skipped: it cancels exactly under BatchNorm.
// --------------------------------------------------------------------------
constexpr int kBtStride = 388;

__global__ __launch_bounds__(256) void gemm_wmma_kernel(
    const float* __restrict__ Wmat,      // (COUT, CIN)
    const float* __restrict__ finterp,   // (B, NL, CH)
    const float* __restrict__ feat_low,  // (B, CL, NL)
    float* __restrict__ y)               // (B, COUT, NL)
{
  __shared__ float Bt[16 * kBtStride];   // Bt[n][c]
  const int b   = blockIdx.z;
  const int n0  = blockIdx.x * 16;
  const int tid = threadIdx.x;

  // f_interp half: contiguous channel reads (coalesced)
  for (int idx = tid; idx < 16 * kCH; idx += 256) {
    int n = idx >> 8;
    int c = idx & 255;
    Bt[n * kBtStride + c] = finterp[((size_t)b * kNL + n0 + n) * kCH + c];
  }
  // feat_low half: contiguous n reads (coalesced 16-float rows)
  for (int idx = tid; idx < 16 * kCL; idx += 256) {
    int c2 = idx >> 4;
    int n  = idx & 15;
    Bt[n * kBtStride + kCH + c2] =
        feat_low[((size_t)b * kCL + c2) * kNL + n0 + n];
  }
  __syncthreads();

  const int wave  = tid >> 5;
  const int lane  = tid & 31;
  const int mt    = blockIdx.y * 8 + wave;   // 0..15
  const int m16   = lane & 15;
  const int khalf = lane >> 4;               // A/B fragments: K = 2*khalf{,+1}

  const float* wrow = Wmat + ((size_t)mt * 16 + m16) * kCIN + 2 * khalf;
  const float* brow = Bt + m16 * kBtStride + 2 * khalf;

  v8f acc = {};
  #pragma unroll 4
  for (int k = 0; k < kCIN; k += 4) {
    v2f a, bf;
    a.x  = wrow[k];
    a.y  = wrow[k + 1];
    bf.x = brow[k];
    bf.y = brow[k + 1];
    acc = __builtin_amdgcn_wmma_f32_16x16x4_f32(
        /*neg_a=*/false, a, /*neg_b=*/false, bf,
        /*c_mod=*/(short)0, acc, /*reuse_a=*/false, /*reuse_b=*/false);
  }

  // D layout: VGPR v, lanes 0-15 -> M=v, lanes 16-31 -> M=v+8; N = lane&15
  const int mbase = mt * 16 + 8 * khalf;
  float* yb = y + (size_t)b * kCOUT * kNL;
  #pragma unroll
  for (int v = 0; v < 8; ++v)
    yb[(size_t)(mbase + v) * kNL + n0 + m16] = acc[v];
}

// --------------------------------------------------------------------------
// Kernel 3: per-channel batch stats (mean, rsqrt(var+eps)) over B*NL.
// --------------------------------------------------------------------------
__global__ __launch_bounds__(256) void bn_stats_kernel(
    const float* __restrict__ y, float* __restrict__ stats)
{
  __shared__ float s1[256];
  __shared__ float s2[256];
  const int o   = blockIdx.x;
  const int tid = threadIdx.x;
  float a = 0.f, q = 0.f;
  for (int i = tid; i < kB * kNL; i += 256) {
    int bb = i >> 13;          // / NL
    int n  = i & (kNL - 1);
    float v = y[((size_t)bb * kCOUT + o) * kNL + n];
    a += v;
    q = fmaf(v, v, q);
  }
  s1[tid] = a; s2[tid] = q;
  __syncthreads();
  for (int off = 128; off > 0; off >>= 1) {
    if (tid < off) { s1[tid] += s1[tid + off]; s2[tid] += s2[tid + off]; }
    __syncthreads();
  }
  if (tid == 0) {
    const float invN = 1.f / (float)(kB * kNL);
    float mean = s1[0] * invN;
    float var  = s2[0] * invN - mean * mean;
    stats[o]         = mean;
    stats[kCOUT + o] = rsqrtf(var + kBnEps);
  }
}

// --------------------------------------------------------------------------
// Kernel 4: in-place normalize + affine + ReLU on y (= d_out).
// --------------------------------------------------------------------------
__global__ __launch_bounds__(256) void bn_relu_kernel(
    float* __restrict__ y, const float* __restrict__ stats,
    const float* __restrict__ gamma, const float* __restrict__ beta)
{
  size_t i = (size_t)blockIdx.x * 256 + threadIdx.x;
  int o = (int)((i >> 13) & (kCOUT - 1));
  float v = y[i];
  v = (v - stats[o]) * stats[kCOUT + o];
  v = fmaf(gamma[o], v, beta[o]);
  y[i] = fmaxf(v, 0.f);
}

extern "C" void kernel_launch(void* const* d_in, const int* in_sizes, int n_in,
                              void* d_out, int out_size, void* d_ws, size_t ws_size,
                              hipStream_t stream) {
  const float* xyz_low   = (const float*)d_in[0];
  const float* xyz_high  = (const float*)d_in[1];
  const float* feat_low  = (const float*)d_in[2];
  const float* feat_high = (const float*)d_in[3];
  const float* Wmat      = (const float*)d_in[4];
  // d_in[5] = conv bias: mathematically cancels under BatchNorm -> unused.
  const float* gamma     = (const float*)d_in[6];
  const float* beta      = (const float*)d_in[7];

  float* out     = (float*)d_out;                        // y, then final result
  float* finterp = (float*)d_ws;                         // (B, NL, CH) = 32 MB
  float* stats   = finterp + (size_t)kB * kNL * kCH;     // 2*COUT floats

  knn_interp_kernel<<<dim3(kNL / 8, kB), 256, 0, stream>>>(
      xyz_low, xyz_high, feat_high, finterp);
  gemm_wmma_kernel<<<dim3(kNL / 16, 2, kB), 256, 0, stream>>>(
      Wmat, finterp, feat_low, out);
  bn_stats_kernel<<<kCOUT, 256, 0, stream>>>(out, stats);
  bn_relu_kernel<<<(int)(((size_t)kB * kCOUT * kNL) / 256), 256, 0, stream>>>(
      out, stats, gamma, beta);
}